// GraphNeuralSolverPBELoss_89936615178647
// MI455X (gfx1250) — compile-verified
//
#include <hip/hip_runtime.h>
#include <stdint.h>

// ---------------------------------------------------------------------------
// GraphNeuralSolver PBE loss, MI455X (gfx1250).
// Memory-bound: ~77MB edge streams @23.3TB/s (~3.3us floor), L2-resident
// node gathers + fp32 atomics. CDNA5 path: double-buffered
// global_load_async_to_lds_b64 prefetch of edge streams (ASYNCcnt),
// split-counter waits, wave32 shuffle reduction.
// ---------------------------------------------------------------------------

#define TPB 256
#define EPT 4                     // edges per thread per tile
#define TILE (TPB * EPT)          // 1024 edges per tile
#define SRC_OFF  0                // LDS layout within one buffer
#define DST_OFF  (TILE * 8)       // 8192
#define ATTR_OFF (TILE * 16)      // 16384
#define BUFBYTES (TILE * 24)      // 24576 per buffer, x2 double buffered

// ---- CDNA5 async global->LDS copy (GV mode), tracked by ASYNCcnt ----------
__device__ __forceinline__ void async_b64(unsigned ldsOff, unsigned long long gaddr) {
  asm volatile("global_load_async_to_lds_b64 %0, %1, off"
               :: "v"(ldsOff), "v"(gaddr) : "memory");
}
__device__ __forceinline__ void wait_async_le(int keep12) {
  if (keep12) asm volatile("s_wait_asynccnt 12" ::: "memory"); // oldest tile (12 ops) done
  else        asm volatile("s_wait_asynccnt 0"  ::: "memory"); // everything done
}
__device__ __forceinline__ void wait_ds0() {
  asm volatile("s_wait_dscnt 0" ::: "memory"); // LDS reads landed before async overwrite
}

__device__ __forceinline__ void atomic_add_f32(float* p, float v) {
  __hip_atomic_fetch_add(p, v, __ATOMIC_RELAXED, __HIP_MEMORY_SCOPE_AGENT);
}

// ---------------------------------------------------------------------------
// Kernel A: node prep. tp = where(mask, pred, target); pack {vm,va} as float2
// (halves gather traffic per edge), precompute {pg-pd, qg-qd}, zero the
// per-node flow accumulators and the scalar output.
// ---------------------------------------------------------------------------
__global__ __launch_bounds__(256) void pbe_node_prep(
    const float* __restrict__ pred, const float* __restrict__ target,
    const unsigned char* __restrict__ mask,
    float2* __restrict__ vmva, float2* __restrict__ imb0,
    float2* __restrict__ calc, float* __restrict__ out, int n_nodes)
{
  int i = blockIdx.x * blockDim.x + threadIdx.x;
  if (i == 0) out[0] = 0.0f;                      // must rewrite d_out every call
  if (i >= n_nodes) return;
  float tp[6];
#pragma unroll
  for (int c = 0; c < 6; ++c) {
    float p = pred[i * 6 + c], t = target[i * 6 + c];
    tp[c] = mask[i * 6 + c] ? p : t;
  }
  vmva[i] = make_float2(tp[0], tp[1]);            // VM, VA
  imb0[i] = make_float2(tp[2] - tp[4], tp[3] - tp[5]); // PG-PD, QG-QD
  calc[i] = make_float2(0.0f, 0.0f);
}

// ---------------------------------------------------------------------------
// Kernel B: edge pipeline. Each thread owns edge slots {t, t+256, t+512,
// t+768} of a 1024-edge tile, so each async_b64 is lane-contiguous (2KB per
// 256-lane issue) and every thread reads back exactly the LDS it loaded ->
// no barriers, only counter waits. 12 async ops per thread per tile.
// ---------------------------------------------------------------------------
__device__ __forceinline__ void issue_tile(
    const long long* src64, const long long* dst64, const float* attr,
    long long tileIdx, unsigned bufBase)
{
  long long eb = tileIdx * (long long)TILE;
#pragma unroll
  for (int j = 0; j < EPT; ++j) {
    unsigned slot = threadIdx.x + j * TPB;
    unsigned lo = bufBase + slot * 8u;
    async_b64(lo + SRC_OFF,  (unsigned long long)(uintptr_t)(src64 + eb + slot));
    async_b64(lo + DST_OFF,  (unsigned long long)(uintptr_t)(dst64 + eb + slot));
    async_b64(lo + ATTR_OFF, (unsigned long long)(uintptr_t)(attr + 2 * (eb + slot)));
  }
}

__global__ __launch_bounds__(TPB) void pbe_edge_kernel(
    const long long* __restrict__ src64, const long long* __restrict__ dst64,
    const float* __restrict__ attr, const float2* __restrict__ vmva,
    float2* __restrict__ calc, long long n_edges)
{
  extern __shared__ unsigned char smem[];         // 2 * BUFBYTES, dynamic (base off 0)
  const long long tilesTotal = n_edges / TILE;
  const long long stride = gridDim.x;
  long long k = blockIdx.x;
  int nq = 0;

  if (k < tilesTotal) {                           // prologue: prime both buffers
    issue_tile(src64, dst64, attr, k, 0u);
    nq = 1;
    if (k + stride < tilesTotal) {
      issue_tile(src64, dst64, attr, k + stride, (unsigned)BUFBYTES);
      nq = 2;
    }
  }

  int buf = 0;
  while (k < tilesTotal) {
    wait_async_le(nq > 1);                        // oldest tile's 12 ops complete
    unsigned base = (unsigned)buf * (unsigned)BUFBYTES;
    const unsigned* ls = (const unsigned*)(smem + base + SRC_OFF);  // int64 low words
    const unsigned* ld = (const unsigned*)(smem + base + DST_OFF);
    const float*    la = (const float*)(smem + base + ATTR_OFF);
#pragma unroll
    for (int j = 0; j < EPT; ++j) {
      unsigned slot = threadIdx.x + j * TPB;
      unsigned s = ls[2 * slot];                  // indices < 2^31: low dword suffices
      unsigned d = ld[2 * slot];
      float g  = la[2 * slot];
      float bb = la[2 * slot + 1];
      float2 fs = vmva[s];                        // L2-resident gathers (800KB table)
      float2 ft = vmva[d];
      float ang = fs.y - ft.y;
      float sn, cn;
      sincosf(ang, &sn, &cn);
      float vv = fs.x * ft.x;
      float pf = vv * (g * cn + bb * sn);
      float qf = vv * (g * sn - bb * cn);
      atomic_add_f32(&calc[s].x, pf);             // segment_sum by src
      atomic_add_f32(&calc[s].y, qf);
    }
    nq--;
    long long k2 = k + 2 * stride;
    if (k2 < tilesTotal) {
      wait_ds0();                                 // WAR: LDS reads done before overwrite
      issue_tile(src64, dst64, attr, k2, base);
      nq++;
    }
    buf ^= 1;
    k += stride;
  }

  // Ragged tail (empty for 3.2M edges, kept for generality): plain path.
  long long tail = tilesTotal * (long long)TILE;
  for (long long e = tail + (long long)blockIdx.x * TPB + threadIdx.x;
       e < n_edges; e += (long long)gridDim.x * TPB) {
    unsigned s = (unsigned)src64[e];
    unsigned d = (unsigned)dst64[e];
    float g = attr[2 * e], bb = attr[2 * e + 1];
    float2 fs = vmva[s], ft = vmva[d];
    float ang = fs.y - ft.y;
    float sn, cn; sincosf(ang, &sn, &cn);
    float vv = fs.x * ft.x;
    atomic_add_f32(&calc[s].x, vv * (g * cn + bb * sn));
    atomic_add_f32(&calc[s].y, vv * (g * sn - bb * cn));
  }
}

// ---------------------------------------------------------------------------
// Kernel C: loss = mean(p_imb^2) + mean(q_imb^2). Wave32 shuffle reduction,
// one atomic per block.
// ---------------------------------------------------------------------------
__global__ __launch_bounds__(256) void pbe_reduce(
    const float2* __restrict__ imb0, const float2* __restrict__ calc,
    float* __restrict__ out, int n_nodes)
{
  float acc = 0.0f;
  for (int i = blockIdx.x * blockDim.x + threadIdx.x; i < n_nodes;
       i += gridDim.x * blockDim.x) {
    float2 b0 = imb0[i];
    float2 c  = calc[i];
    float p = b0.x - c.x;
    float q = b0.y - c.y;
    acc += p * p + q * q;
  }
#pragma unroll
  for (int off = 16; off > 0; off >>= 1) acc += __shfl_xor(acc, off, 32);
  __shared__ float ws[8];
  int lane = threadIdx.x & 31, wid = threadIdx.x >> 5;
  if (lane == 0) ws[wid] = acc;
  __syncthreads();
  if (wid == 0) {
    acc = (lane < 8) ? ws[lane] : 0.0f;
#pragma unroll
    for (int off = 4; off > 0; off >>= 1) acc += __shfl_xor(acc, off, 32);
    if (lane == 0) atomic_add_f32(out, acc * (1.0f / (float)n_nodes));
  }
}

// ---------------------------------------------------------------------------
extern "C" void kernel_launch(void* const* d_in, const int* in_sizes, int n_in,
                              void* d_out, int out_size, void* d_ws, size_t ws_size,
                              hipStream_t stream)
{
  const float*         pred   = (const float*)d_in[0];
  const float*         target = (const float*)d_in[1];
  const long long*     eidx   = (const long long*)d_in[2];   // (2, E) int64
  const float*         eattr  = (const float*)d_in[3];       // (E, 2) f32
  const unsigned char* mask   = (const unsigned char*)d_in[4]; // (N, 6) bool (1B)
  float* out = (float*)d_out;

  const int       n_nodes = in_sizes[0] / 6;
  const long long n_edges = (long long)in_sizes[2] / 2;

  // Workspace: vmva | imb0 | calc, each float2[n_nodes] (2.4MB total)
  float2* vmva = (float2*)d_ws;
  float2* imb0 = vmva + n_nodes;
  float2* calc = imb0 + n_nodes;

  const long long* src64 = eidx;
  const long long* dst64 = eidx + n_edges;

  pbe_node_prep<<<(n_nodes + 255) / 256, 256, 0, stream>>>(
      pred, target, mask, vmva, imb0, calc, out, n_nodes);

  pbe_edge_kernel<<<512, TPB, 2 * BUFBYTES, stream>>>(
      src64, dst64, eattr, vmva, calc, n_edges);

  pbe_reduce<<<256, 256, 0, stream>>>(imb0, calc, out, n_nodes);
}